// Discriminator_41618233098577
// MI455X (gfx1250) — compile-verified
//
#include <hip/hip_runtime.h>

typedef __attribute__((ext_vector_type(2))) float v2f;
typedef __attribute__((ext_vector_type(8))) float v8f;

#define FD 128           // feature dim (D == H == 128)
#define WSLICE (FD * FD) // one hop-weight slice, 16384 floats

// ---------------------------------------------------------------- utilities

// deg[row[e]] += 1
__global__ void deg_kernel(const long long* __restrict__ row,
                           float* __restrict__ deg, int E) {
  int e = blockIdx.x * blockDim.x + threadIdx.x;
  if (e < E)
    __hip_atomic_fetch_add(&deg[(size_t)row[e]], 1.0f,
                           __ATOMIC_RELAXED, __HIP_MEMORY_SCOPE_AGENT);
}

// deg -> deg^{-1/2} (0 where deg == 0), in place
__global__ void dinv_kernel(float* __restrict__ deg, int N) {
  int i = blockIdx.x * blockDim.x + threadIdx.x;
  if (i < N) {
    float d = deg[i];
    deg[i] = (d > 0.0f) ? rsqrtf(d) : 0.0f;
  }
}

// norm[e] = dinv[row[e]] * dinv[col[e]]
__global__ void norm_kernel(const long long* __restrict__ row,
                            const long long* __restrict__ col,
                            const float* __restrict__ dinv,
                            float* __restrict__ norm, int E) {
  int e = blockIdx.x * blockDim.x + threadIdx.x;
  if (e < E) norm[e] = dinv[(size_t)row[e]] * dinv[(size_t)col[e]];
}

// ------------------------------------------------- sparse propagation h<-Âh
// One wave32 per edge: 32 lanes x float4 = 128 features.
// Edge index forced wave-uniform via readfirstlane so row/col/norm become
// scalar (SMEM) loads. Gather is one coalesced 512B line (L2-resident:
// the 25.6MB feature matrix fits in the 192MB L2); scatter uses relaxed
// agent-scope float atomics (global_atomic_add_f32, no-return / STOREcnt).
__global__ void prop_kernel(const long long* __restrict__ row,
                            const long long* __restrict__ col,
                            const float* __restrict__ norm,
                            const float* __restrict__ hin,
                            float* __restrict__ hout, int E) {
  int e = __builtin_amdgcn_readfirstlane(
      (int)((blockIdx.x * blockDim.x + threadIdx.x) >> 5));
  int lane = threadIdx.x & 31;
  if (e >= E) return;
  long long r = row[e];
  long long c = col[e];
  float w = norm[e];
  const float4* src = (const float4*)(hin + (size_t)c * FD);
  float4 v = src[lane];
  float* dst = hout + (size_t)r * FD + lane * 4;
  __hip_atomic_fetch_add(dst + 0, v.x * w, __ATOMIC_RELAXED, __HIP_MEMORY_SCOPE_AGENT);
  __hip_atomic_fetch_add(dst + 1, v.y * w, __ATOMIC_RELAXED, __HIP_MEMORY_SCOPE_AGENT);
  __hip_atomic_fetch_add(dst + 2, v.z * w, __ATOMIC_RELAXED, __HIP_MEMORY_SCOPE_AGENT);
  __hip_atomic_fetch_add(dst + 3, v.w * w, __ATOMIC_RELAXED, __HIP_MEMORY_SCOPE_AGENT);
}

// ------------------------------------------------------- WMMA fp32 GEMM
// acc[N x 128] (+)= hin[N x 128] @ Wk[128 x 128], bias-init when first != 0.
// Block = 256 threads = 8 wave32; wave w owns rows [blk*128 + w*16, +16).
// W slice (64KB) staged into LDS with gfx1250 async memory->LDS loads
// (ASYNCcnt path), then K swept in 32 chunks of 4 via V_WMMA_F32_16X16X4_F32.
//
// fp32 16x4 A fragment per lane (v2f): lanes 0-15 hold K={0,1} of row M=lane,
// lanes 16-31 hold K={2,3} of row M=lane-16. B (4x16) mirrors that striping.
// C/D v8f: VGPR j holds (M=j, lanes 0-15) and (M=j+8, lanes 16-31), N=lane&15.
__global__ void gemm_tag_kernel(const float* __restrict__ hin,
                                const float* __restrict__ Wk,
                                const float* __restrict__ bias,
                                float* __restrict__ acc,
                                int N, int first) {
  __shared__ __align__(16) float Wlds[WSLICE];
  {
    // 4096 x 16B lines, 256 threads -> 16 async b128 loads per thread
    unsigned ldsbase = (unsigned)(uintptr_t)(&Wlds[0]);
    for (int i = threadIdx.x; i < WSLICE / 4; i += 256) {
      unsigned loff = ldsbase + (unsigned)i * 16u;
      const float* gptr = Wk + (size_t)i * 4;
      asm volatile("global_load_async_to_lds_b128 %0, %1, off"
                   :: "v"(loff), "v"(gptr)
                   : "memory");
    }
    asm volatile("s_wait_asynccnt 0" ::: "memory");
  }
  __syncthreads();

  const int wave = threadIdx.x >> 5;
  const int lane = threadIdx.x & 31;
  const int row0 = blockIdx.x * 128 + wave * 16;
  if (row0 >= N) return;  // whole-wave exit (N multiple of 16); barrier already passed

  const int m = lane & 15;   // row (A/C) or col (B/C) index within tile
  const int kh = lane >> 4;  // K-half select

  v8f c[8];
  if (first) {
    for (int nt = 0; nt < 8; ++nt) {
      float bv = bias[nt * 16 + m];
      for (int j = 0; j < 8; ++j) c[nt][j] = bv;
    }
  } else {
    for (int nt = 0; nt < 8; ++nt)
      for (int j = 0; j < 8; ++j)
        c[nt][j] = acc[(size_t)(row0 + j + kh * 8) * FD + nt * 16 + m];
  }

  for (int kc = 0; kc < 32; ++kc) {
    const int k0 = kc * 4;
    const float* ap = hin + (size_t)(row0 + m) * FD + k0 + kh * 2;
    v2f a;
    a[0] = ap[0];
    a[1] = ap[1];
    for (int nt = 0; nt < 8; ++nt) {
      const int n = nt * 16 + m;
      v2f b;
      b[0] = Wlds[(k0 + kh * 2 + 0) * FD + n];
      b[1] = Wlds[(k0 + kh * 2 + 1) * FD + n];
      c[nt] = __builtin_amdgcn_wmma_f32_16x16x4_f32(
          /*neg_a=*/false, a, /*neg_b=*/false, b,
          /*c_mod=*/(short)0, c[nt], /*reuse_a=*/false, /*reuse_b=*/false);
    }
  }

  for (int nt = 0; nt < 8; ++nt)
    for (int j = 0; j < 8; ++j)
      acc[(size_t)(row0 + j + kh * 8) * FD + nt * 16 + m] = c[nt][j];
}

// ---------------------------------------------------------------- PReLU
__global__ void prelu_kernel(const float* __restrict__ x,
                             const float* __restrict__ a,
                             float* __restrict__ y, long n) {
  long i = (long)blockIdx.x * blockDim.x + threadIdx.x;
  if (i < n) {
    float v = x[i];
    float al = a[0];
    y[i] = (v >= 0.0f) ? v : al * v;
  }
}

// ------------------------------------------------- global add pool -> g[128]
__global__ void pool_kernel(const float* __restrict__ h,
                            float* __restrict__ g, int N) {
  int t = threadIdx.x;  // feature 0..127
  float s = 0.0f;
  for (int n = blockIdx.x; n < N; n += gridDim.x)
    s += h[(size_t)n * FD + t];
  __hip_atomic_fetch_add(&g[t], s, __ATOMIC_RELAXED, __HIP_MEMORY_SCOPE_AGENT);
}

// ------------------------------------------------- head: out = g.Wout + bout
__global__ void head_kernel(const float* __restrict__ g,
                            const float* __restrict__ Wout,
                            const float* __restrict__ bout,
                            float* __restrict__ out) {
  __shared__ float red[128];
  int t = threadIdx.x;
  red[t] = g[t] * Wout[t];
  __syncthreads();
  for (int s = 64; s > 0; s >>= 1) {
    if (t < s) red[t] += red[t + s];
    __syncthreads();
  }
  if (t == 0) out[0] = red[0] + bout[0];
}

// ================================================================== launch

extern "C" void kernel_launch(void* const* d_in, const int* in_sizes, int n_in,
                              void* d_out, int out_size, void* d_ws, size_t ws_size,
                              hipStream_t stream) {
  const float* x        = (const float*)d_in[0];
  const long long* ei   = (const long long*)d_in[1];  // int64 (2, E)
  const float* W1       = (const float*)d_in[2];      // (4,128,128)
  const float* b1       = (const float*)d_in[3];
  const float* a1       = (const float*)d_in[4];
  const float* W2       = (const float*)d_in[5];
  const float* b2       = (const float*)d_in[6];
  const float* a2       = (const float*)d_in[7];
  const float* Wout     = (const float*)d_in[8];
  const float* bout     = (const float*)d_in[9];
  float* out            = (float*)d_out;

  const int N = in_sizes[0] / FD;
  const int E = in_sizes[1] / 2;
  const long NF = (long)N * FD;

  const long long* row = ei;
  const long long* col = ei + E;

  // workspace carve-up (floats)
  float* ws    = (float*)d_ws;
  float* deg   = ws;                // N   (becomes dinv)
  float* norm  = deg + N;           // E
  float* bufA  = norm + E;          // N*FD
  float* bufB  = bufA + NF;         // N*FD
  float* acc   = bufB + NF;         // N*FD
  float* g     = acc + NF;          // FD

  const int TB = 256;
  dim3 blkE((E + TB - 1) / TB);
  dim3 blkN((N + TB - 1) / TB);
  dim3 blkNF((NF + TB - 1) / TB);
  dim3 blkProp(((long)E * 32 + TB - 1) / TB);   // one wave32 per edge
  dim3 blkGemm((N + 127) / 128);

  // --- normalization
  hipMemsetAsync(deg, 0, (size_t)N * sizeof(float), stream);
  deg_kernel<<<blkE, TB, 0, stream>>>(row, deg, E);
  dinv_kernel<<<blkN, TB, 0, stream>>>(deg, N);
  norm_kernel<<<blkE, TB, 0, stream>>>(row, col, deg, norm, E);

  // --- layer 1: acc = sum_k (Â^k x) W1[k] + b1
  gemm_tag_kernel<<<blkGemm, TB, 0, stream>>>(x, W1 + 0 * WSLICE, b1, acc, N, 1);
  hipMemsetAsync(bufB, 0, (size_t)NF * sizeof(float), stream);
  prop_kernel<<<blkProp, TB, 0, stream>>>(row, col, norm, x, bufB, E);
  gemm_tag_kernel<<<blkGemm, TB, 0, stream>>>(bufB, W1 + 1 * WSLICE, b1, acc, N, 0);
  hipMemsetAsync(bufA, 0, (size_t)NF * sizeof(float), stream);
  prop_kernel<<<blkProp, TB, 0, stream>>>(row, col, norm, bufB, bufA, E);
  gemm_tag_kernel<<<blkGemm, TB, 0, stream>>>(bufA, W1 + 2 * WSLICE, b1, acc, N, 0);
  hipMemsetAsync(bufB, 0, (size_t)NF * sizeof(float), stream);
  prop_kernel<<<blkProp, TB, 0, stream>>>(row, col, norm, bufA, bufB, E);
  gemm_tag_kernel<<<blkGemm, TB, 0, stream>>>(bufB, W1 + 3 * WSLICE, b1, acc, N, 0);
  prelu_kernel<<<blkNF, TB, 0, stream>>>(acc, a1, bufA, NF);  // bufA = layer-2 input

  // --- layer 2
  gemm_tag_kernel<<<blkGemm, TB, 0, stream>>>(bufA, W2 + 0 * WSLICE, b2, acc, N, 1);
  hipMemsetAsync(bufB, 0, (size_t)NF * sizeof(float), stream);
  prop_kernel<<<blkProp, TB, 0, stream>>>(row, col, norm, bufA, bufB, E);
  gemm_tag_kernel<<<blkGemm, TB, 0, stream>>>(bufB, W2 + 1 * WSLICE, b2, acc, N, 0);
  hipMemsetAsync(bufA, 0, (size_t)NF * sizeof(float), stream);
  prop_kernel<<<blkProp, TB, 0, stream>>>(row, col, norm, bufB, bufA, E);
  gemm_tag_kernel<<<blkGemm, TB, 0, stream>>>(bufA, W2 + 2 * WSLICE, b2, acc, N, 0);
  hipMemsetAsync(bufB, 0, (size_t)NF * sizeof(float), stream);
  prop_kernel<<<blkProp, TB, 0, stream>>>(row, col, norm, bufA, bufB, E);
  gemm_tag_kernel<<<blkGemm, TB, 0, stream>>>(bufB, W2 + 3 * WSLICE, b2, acc, N, 0);
  prelu_kernel<<<blkNF, TB, 0, stream>>>(acc, a2, bufA, NF);  // final node features

  // --- pool + head
  hipMemsetAsync(g, 0, FD * sizeof(float), stream);
  pool_kernel<<<dim3(256), FD, 0, stream>>>(bufA, g, N);
  head_kernel<<<dim3(1), FD, 0, stream>>>(g, Wout, bout, out);
}